// ARIMAModule_59906203845008
// MI455X (gfx1250) — compile-verified
//
#include <hip/hip_runtime.h>
#include <stdint.h>

// =====================================================================
// ARIMA(4,d,4) batch simulation for MI455X (gfx1250).
// Memory-bound (~203 MB total traffic -> ~9us at 23.3 TB/s).
// Exact chunked linear-scan parallelization over time; CDNA5 async
// global->LDS staging (ASYNCcnt double buffering) for coalescing.
// =====================================================================

#ifndef __has_builtin
#define __has_builtin(x) 0
#endif

#if __has_builtin(__builtin_amdgcn_global_load_async_to_lds_b64)
#define ARIMA_ASYNC 1
// Builtin signature (from hipcc diagnostic): param0 is
// 'v2i __device__*' i.e. int-vector in addrspace(1); param1 is the LDS
// (addrspace(3)) counterpart.
typedef int v2i_t __attribute__((vector_size(2 * sizeof(int))));
typedef __attribute__((address_space(1))) v2i_t gbl_v2i;
typedef __attribute__((address_space(3))) v2i_t lds_v2i;
#else
#define ARIMA_ASYNC 0
#endif

#if __has_builtin(__builtin_amdgcn_s_wait_asynccnt)
#define WAIT_ASYNC(n) __builtin_amdgcn_s_wait_asynccnt(n)
#else
// Portable fallback: assembles on gfx1250, harmless no-op semantics when
// no async ops are outstanding.
#define WAIT_ASYNC(n) asm volatile("s_wait_asynccnt %0" ::"n"(n) : "memory")
#endif

#define SB 128      // series per block (threads per block)
#define TS 32       // time steps per LDS tile
#define NPAD 36     // noise-tile row stride in floats (8B-aligned rows, 2-way banks)
#define OPAD 33     // output-tile row stride in floats (conflict-free)
#define NCHUNK 32   // time chunks (grid parallelism: 32 x N/SB blocks)
#define LOADS_PER_TILE 16  // async b64 instructions issued per thread per tile

// ---------------------------------------------------------------------
// Stage one SB x TS noise tile into LDS. Row base addresses are only
// 8B aligned (total_T = 4146 floats), so use b64 transfers. Uniform
// instruction count per wave (address-clamped, never skipped) so
// ASYNCcnt bookkeeping is exact for double buffering.
// ---------------------------------------------------------------------
__device__ __forceinline__ void load_tile(const float* __restrict__ noise,
                                          float* nb, int n0, long long total_T,
                                          long long tt, int tid) {
#pragma unroll
  for (int j = 0; j < LOADS_PER_TILE; ++j) {
    int q = j * SB + tid;        // [0, SB*TS/2)
    int row = q >> 4;            // 16 b64 slots per row
    int col = (q & 15) << 1;     // float column (even)
    long long t = tt + col;
    if (t > total_T - 2) t = total_T - 2;  // clamp; clamped slots never read
    const float* g = noise + (long long)(n0 + row) * total_T + t;
    float* l = nb + row * NPAD + col;
#if ARIMA_ASYNC
    __builtin_amdgcn_global_load_async_to_lds_b64((gbl_v2i*)g, (lds_v2i*)l, 0,
                                                  0);
#else
    float2 v = *reinterpret_cast<const float2*>(g);
    l[0] = v.x;
    l[1] = v.y;
#endif
  }
}

// ---------------------------------------------------------------------
// Kernel 1: per-series homogeneous propagation over one chunk length L.
// Produces M = A^L (4x4 companion power), u (response of sum(y) to the
// initial state), w (response of sum(cumsum(y))). Layout: H[k*N+n],
// k = 0..15 -> M[row][colbasis], 16..19 -> u, 20..23 -> w.
// ---------------------------------------------------------------------
__global__ void k_homo(const float* __restrict__ phi, int N, int L,
                       float* __restrict__ H) {
  int n = blockIdx.x * blockDim.x + threadIdx.x;
  if (n >= N) return;
  const float p0 = phi[n * 4 + 0], p1 = phi[n * 4 + 1];
  const float p2 = phi[n * 4 + 2], p3 = phi[n * 4 + 3];
  float h[4][4];
#pragma unroll
  for (int j = 0; j < 4; ++j)
#pragma unroll
    for (int k = 0; k < 4; ++k) h[j][k] = (j == k) ? 1.f : 0.f;
  float s1[4] = {0.f, 0.f, 0.f, 0.f};
  float w[4] = {0.f, 0.f, 0.f, 0.f};
  for (int t = 0; t < L; ++t) {
#pragma unroll
    for (int j = 0; j < 4; ++j) {
      float y = p0 * h[j][0] + p1 * h[j][1] + p2 * h[j][2] + p3 * h[j][3];
      s1[j] += y;
      w[j] += s1[j];
      h[j][3] = h[j][2]; h[j][2] = h[j][1]; h[j][1] = h[j][0]; h[j][0] = y;
    }
  }
#pragma unroll
  for (int k = 0; k < 4; ++k)
#pragma unroll
    for (int j = 0; j < 4; ++j)
      H[(long long)(k * 4 + j) * N + n] = h[j][k];  // M[k][j]
#pragma unroll
  for (int j = 0; j < 4; ++j) {
    H[(long long)(16 + j) * N + n] = s1[j];
    H[(long long)(20 + j) * N + n] = w[j];
  }
}

// ---------------------------------------------------------------------
// Kernel 2 (phase A): per (chunk, series) run the recurrence from a zero
// y-history (eps history is exact: it depends only on noise). Record the
// final 4-state f and local sums S1 = sum(y), S2 = sum(cumsum(y)).
// FS planes: [0..3]=f, [4]=S1, [5]=S2; plane stride NCHUNK*N, idx c*N+n.
// ---------------------------------------------------------------------
__global__ __launch_bounds__(SB) void k_phaseA(
    const float* __restrict__ phi, const float* __restrict__ theta,
    const float* __restrict__ icpt, const float* __restrict__ sig,
    const float* __restrict__ noise, int N, long long total_T, int L,
    float* __restrict__ FS) {
  __shared__ float nbuf[2][SB * NPAD];
  const int tid = threadIdx.x;
  const int n0 = blockIdx.x * SB;
  const int c = blockIdx.y;
  const int n = n0 + tid;
  const long long t0 = (long long)c * L;
  const long long rem = total_T - t0;
  if (rem <= 0) return;  // block-uniform
  const int len = rem < (long long)L ? (int)rem : L;
  const int nt = (len + TS - 1) / TS;

  const float p0 = phi[n * 4 + 0], p1 = phi[n * 4 + 1];
  const float p2 = phi[n * 4 + 2], p3 = phi[n * 4 + 3];
  const float q0 = theta[n * 4 + 0], q1 = theta[n * 4 + 1];
  const float q2 = theta[n * 4 + 2], q3 = theta[n * 4 + 3];
  const float c0 = icpt[n];
  const float sg = sig[n];

  const long long nrow = (long long)n * total_T;
  float eh0 = (t0 >= 1) ? noise[nrow + t0 - 1] * sg : 0.f;
  float eh1 = (t0 >= 2) ? noise[nrow + t0 - 2] * sg : 0.f;
  float eh2 = (t0 >= 3) ? noise[nrow + t0 - 3] * sg : 0.f;
  float eh3 = (t0 >= 4) ? noise[nrow + t0 - 4] * sg : 0.f;

  float yh0 = 0.f, yh1 = 0.f, yh2 = 0.f, yh3 = 0.f;
  float r1 = 0.f, r2 = 0.f;

  load_tile(noise, nbuf[0], n0, total_T, t0, tid);
  for (int si = 0; si < nt; ++si) {
    if (si + 1 < nt) {
      load_tile(noise, nbuf[(si + 1) & 1], n0, total_T,
                t0 + (long long)(si + 1) * TS, tid);
      WAIT_ASYNC(LOADS_PER_TILE);  // current tile done, next in flight
    } else {
      WAIT_ASYNC(0);
    }
    __syncthreads();
    const float* nb = nbuf[si & 1] + tid * NPAD;
    const int base = si * TS;
    const int m = (len - base) < TS ? (len - base) : TS;
    for (int i = 0; i < m; ++i) {
      float eps = nb[i] * sg;
      float y = c0 + p0 * yh0 + p1 * yh1 + p2 * yh2 + p3 * yh3 + eps +
                q0 * eh0 + q1 * eh1 + q2 * eh2 + q3 * eh3;
      r1 += y;
      r2 += r1;
      yh3 = yh2; yh2 = yh1; yh1 = yh0; yh0 = y;
      eh3 = eh2; eh2 = eh1; eh1 = eh0; eh0 = eps;
    }
    __syncthreads();  // protect nbuf before next prefetch overwrites it
  }
  const long long CN = (long long)NCHUNK * N;
  const long long s = (long long)c * N + n;
  FS[0 * CN + s] = yh0; FS[1 * CN + s] = yh1;
  FS[2 * CN + s] = yh2; FS[3 * CN + s] = yh3;
  FS[4 * CN + s] = r1;  FS[5 * CN + s] = r2;
}

// ---------------------------------------------------------------------
// Kernel 3: per-series 32-step scan over chunks. Exact initial states
// s_c and cumsum carries via linearity: s_{c+1} = M s_c + f_c,
// S1exact = S1loc + u.s, S2exact = S2loc + w.s,
// cc2' = cc2 + S2exact + cc1*L, cc1' = cc1 + S1exact.
// SC planes: [0..3]=s, [4]=cc1, [5]=cc2.
// ---------------------------------------------------------------------
__global__ void k_scan(const float* __restrict__ H, const float* __restrict__ FS,
                       float* __restrict__ SC, int N, int L) {
  int n = blockIdx.x * blockDim.x + threadIdx.x;
  if (n >= N) return;
  float M[4][4], u[4], w[4];
#pragma unroll
  for (int k = 0; k < 4; ++k)
#pragma unroll
    for (int j = 0; j < 4; ++j) M[k][j] = H[(long long)(k * 4 + j) * N + n];
#pragma unroll
  for (int j = 0; j < 4; ++j) {
    u[j] = H[(long long)(16 + j) * N + n];
    w[j] = H[(long long)(20 + j) * N + n];
  }
  const long long CN = (long long)NCHUNK * N;
  float s0 = 0.f, s1v = 0.f, s2 = 0.f, s3 = 0.f, cc1 = 0.f, cc2 = 0.f;
  for (int c = 0; c < NCHUNK; ++c) {
    const long long s = (long long)c * N + n;
    SC[0 * CN + s] = s0;  SC[1 * CN + s] = s1v;
    SC[2 * CN + s] = s2;  SC[3 * CN + s] = s3;
    SC[4 * CN + s] = cc1; SC[5 * CN + s] = cc2;
    float f0 = FS[0 * CN + s], f1 = FS[1 * CN + s];
    float f2 = FS[2 * CN + s], f3 = FS[3 * CN + s];
    float S1l = FS[4 * CN + s], S2l = FS[5 * CN + s];
    float S1e = S1l + u[0] * s0 + u[1] * s1v + u[2] * s2 + u[3] * s3;
    float S2e = S2l + w[0] * s0 + w[1] * s1v + w[2] * s2 + w[3] * s3;
    cc2 += S2e + cc1 * (float)L;
    cc1 += S1e;
    float a0 = f0 + M[0][0] * s0 + M[0][1] * s1v + M[0][2] * s2 + M[0][3] * s3;
    float a1 = f1 + M[1][0] * s0 + M[1][1] * s1v + M[1][2] * s2 + M[1][3] * s3;
    float a2 = f2 + M[2][0] * s0 + M[2][1] * s1v + M[2][2] * s2 + M[2][3] * s3;
    float a3 = f3 + M[3][0] * s0 + M[3][1] * s1v + M[3][2] * s2 + M[3][3] * s3;
    s0 = a0; s1v = a1; s2 = a2; s3 = a3;
  }
}

// ---------------------------------------------------------------------
// Kernel 4 (phase C): re-run each chunk from its exact state, stream the
// first/second prefix sums with carries, select by d, stage the output
// tile in LDS and store fully coalesced.
// ---------------------------------------------------------------------
__global__ __launch_bounds__(SB) void k_phaseC(
    const float* __restrict__ phi, const float* __restrict__ theta,
    const float* __restrict__ icpt, const float* __restrict__ sig,
    const float* __restrict__ noise, const int* __restrict__ dvals,
    const float* __restrict__ SC, int N, long long total_T, int L, int burn,
    int T_out, float* __restrict__ out) {
  __shared__ float nbuf[2][SB * NPAD];
  __shared__ float obuf[SB * OPAD];
  const int tid = threadIdx.x;
  const int n0 = blockIdx.x * SB;
  const int c = blockIdx.y;
  const int n = n0 + tid;
  const long long t0 = (long long)c * L;
  const long long rem = total_T - t0;
  if (rem <= 0) return;  // block-uniform
  const int len = rem < (long long)L ? (int)rem : L;
  const int nt = (len + TS - 1) / TS;

  const float p0 = phi[n * 4 + 0], p1 = phi[n * 4 + 1];
  const float p2 = phi[n * 4 + 2], p3 = phi[n * 4 + 3];
  const float q0 = theta[n * 4 + 0], q1 = theta[n * 4 + 1];
  const float q2 = theta[n * 4 + 2], q3 = theta[n * 4 + 3];
  const float c0 = icpt[n];
  const float sg = sig[n];
  const int d = dvals[n];

  const long long CN = (long long)NCHUNK * N;
  const long long sidx = (long long)c * N + n;
  float yh0 = SC[0 * CN + sidx], yh1 = SC[1 * CN + sidx];
  float yh2 = SC[2 * CN + sidx], yh3 = SC[3 * CN + sidx];
  const float cc1 = SC[4 * CN + sidx], cc2 = SC[5 * CN + sidx];

  const long long nrow = (long long)n * total_T;
  float eh0 = (t0 >= 1) ? noise[nrow + t0 - 1] * sg : 0.f;
  float eh1 = (t0 >= 2) ? noise[nrow + t0 - 2] * sg : 0.f;
  float eh2 = (t0 >= 3) ? noise[nrow + t0 - 3] * sg : 0.f;
  float eh3 = (t0 >= 4) ? noise[nrow + t0 - 4] * sg : 0.f;

  float r1 = 0.f, r2 = 0.f;

  load_tile(noise, nbuf[0], n0, total_T, t0, tid);
  for (int si = 0; si < nt; ++si) {
    if (si + 1 < nt) {
      load_tile(noise, nbuf[(si + 1) & 1], n0, total_T,
                t0 + (long long)(si + 1) * TS, tid);
      WAIT_ASYNC(LOADS_PER_TILE);
    } else {
      WAIT_ASYNC(0);
    }
    __syncthreads();
    const float* nb = nbuf[si & 1] + tid * NPAD;
    const int base = si * TS;
    const int m = (len - base) < TS ? (len - base) : TS;
    for (int i = 0; i < m; ++i) {
      float eps = nb[i] * sg;
      float y = c0 + p0 * yh0 + p1 * yh1 + p2 * yh2 + p3 * yh3 + eps +
                q0 * eh0 + q1 * eh1 + q2 * eh2 + q3 * eh3;
      r1 += y;
      float c1g = cc1 + r1;   // first-order cumsum (global)
      r2 += c1g;
      float c2g = cc2 + r2;   // second-order cumsum (global)
      float v = (d <= 0) ? y : ((d == 1) ? c1g : c2g);
      obuf[tid * OPAD + i] = v;
      yh3 = yh2; yh2 = yh1; yh1 = yh0; yh0 = y;
      eh3 = eh2; eh2 = eh1; eh1 = eh0; eh0 = eps;
    }
    __syncthreads();
    // Coalesced store: lane -> consecutive output column.
    const long long tt = t0 + (long long)base;
#pragma unroll
    for (int j = 0; j < TS; ++j) {
      int qq = j * SB + tid;
      int row = qq >> 5;
      int col = qq & 31;
      long long t = tt + col;
      int gi = (int)(t - t0);
      if (gi < len && t >= (long long)burn)
        out[(long long)(n0 + row) * T_out + (t - burn)] =
            obuf[row * OPAD + col];
    }
    __syncthreads();  // obuf + nbuf reuse
  }
}

// ---------------------------------------------------------------------
// Host launcher. Inputs (setup_inputs order): phi(N,4) theta(N,4)
// intercepts(N) sigmas(N) noise(N,total_T) d_vals(N) T burn_in.
// Workspace: H (24*N) | FS (6*NCHUNK*N) | SC (6*NCHUNK*N) floats = ~6.4MB.
// ---------------------------------------------------------------------
extern "C" void kernel_launch(void* const* d_in, const int* in_sizes, int n_in,
                              void* d_out, int out_size, void* d_ws,
                              size_t ws_size, hipStream_t stream) {
  (void)n_in; (void)ws_size;
  const float* phi = (const float*)d_in[0];
  const float* theta = (const float*)d_in[1];
  const float* icpt = (const float*)d_in[2];
  const float* sig = (const float*)d_in[3];
  const float* noise = (const float*)d_in[4];
  const int* dvals = (const int*)d_in[5];

  const int N = in_sizes[2];                       // 4096
  const long long total_T = in_sizes[4] / N;       // 4146
  const int T_out = out_size / N;                  // 4096
  const int burn = (int)(total_T - T_out);         // 50
  const int L = (int)((total_T + NCHUNK - 1) / NCHUNK);  // 130

  float* H = (float*)d_ws;
  float* FS = H + (size_t)24 * N;
  float* SC = FS + (size_t)6 * NCHUNK * N;

  k_homo<<<(N + 255) / 256, 256, 0, stream>>>(phi, N, L, H);
  dim3 grid(N / SB, NCHUNK);
  k_phaseA<<<grid, SB, 0, stream>>>(phi, theta, icpt, sig, noise, N, total_T,
                                    L, FS);
  k_scan<<<(N + 255) / 256, 256, 0, stream>>>(H, FS, SC, N, L);
  k_phaseC<<<grid, SB, 0, stream>>>(phi, theta, icpt, sig, noise, dvals, SC, N,
                                    total_T, L, burn, T_out, (float*)d_out);
}